// AxialAttention_13469017440501
// MI455X (gfx1250) — compile-verified
//
#include <hip/hip_runtime.h>
#include <hip/hip_bf16.h>
#include <cstddef>

// ---------------------------------------------------------------------------
// AxialAttention on MI455X (gfx1250), fp32 throughout (bandwidth-bound problem,
// so full precision is free). Matrix work uses V_WMMA_F32_16X16X4_F32.
// Two-pass "flash" attention avoids materializing the 3x134MB logit tensors.
// Cross-lane reductions use single-instruction ds_swizzle (xor-within-32).
// ---------------------------------------------------------------------------

typedef float  v2f __attribute__((ext_vector_type(2)));
typedef float  v8f __attribute__((ext_vector_type(8)));

#define WMMA_F32_4(a, b, c) \
  __builtin_amdgcn_wmma_f32_16x16x4_f32(false, (a), false, (b), (short)0, (c), false, false)

// ds_swizzle_b32 group-of-32 xor pattern: offset[4:0]=and=0x1F, [9:5]=or=0,
// [14:10]=xor mask. Requires full EXEC (true at all our reduction sites:
// WMMA kernels keep EXEC all-ones).
template <int XMASK>
__device__ __forceinline__ float swzXor(float v) {
  return __int_as_float(
      __builtin_amdgcn_ds_swizzle(__float_as_int(v), (XMASK << 10) | 0x1F));
}

__device__ __forceinline__ float redAdd16(float v) {
  v += swzXor<1>(v);
  v += swzXor<2>(v);
  v += swzXor<4>(v);
  v += swzXor<8>(v);
  return v;
}
__device__ __forceinline__ float redAdd32(float v) {
  v = redAdd16(v);
  v += swzXor<16>(v);
  return v;
}
__device__ __forceinline__ float redMax16(float v) {
  v = fmaxf(v, swzXor<1>(v));
  v = fmaxf(v, swzXor<2>(v));
  v = fmaxf(v, swzXor<4>(v));
  v = fmaxf(v, swzXor<8>(v));
  return v;
}

// ---------------------------------------------------------------------------
// K1: QKV projection GEMM. Stacked weights (q:0..63, k:64..127, v:128..255)
// times x[b, :, hw]. One wave = one 16x16 output tile, K=128 in 4-wide steps
// on the fp32 matrix pipe. Also accumulates per-channel sum/sumsq (BN stats).
// qkv layout: [b][o][hw]  (o in 0..255, hw = h*128+w)
// ---------------------------------------------------------------------------
__global__ void k_gemm_qkv(const float* __restrict__ x,
                           const float* __restrict__ wq,
                           const float* __restrict__ wk,
                           const float* __restrict__ wv,
                           float* __restrict__ qkv,
                           float* __restrict__ ch_sum,
                           float* __restrict__ ch_ss) {
  const int lane = threadIdx.x & 31;
  const int wave = threadIdx.x >> 5;
  const int tile = blockIdx.x * 8 + wave;   // 32768 tiles total
  const int b    = tile >> 14;              // 16384 tiles per batch
  const int rem  = tile & 16383;
  const int mT   = rem >> 10;               // 16 channel tiles
  const int nT   = rem & 1023;              // 1024 position tiles
  const int hl   = lane >> 4;               // half-wave (K split / M split)
  const int ln   = lane & 15;

  const int m  = mT * 16 + ln;              // A-operand row (output channel)
  const int hw = nT * 16 + ln;              // B/D-operand column (position)

  const float* wrow;
  if (m < 64)       wrow = wq + (size_t)m * 128;
  else if (m < 128) wrow = wk + (size_t)(m - 64) * 128;
  else              wrow = wv + (size_t)(m - 128) * 128;

  const float* xb = x + (size_t)b * 128 * 16384 + hw;

  v8f acc = {0.f, 0.f, 0.f, 0.f, 0.f, 0.f, 0.f, 0.f};
  for (int kk = 0; kk < 128; kk += 4) {
    const int kb = kk + hl * 2;
    v2f a, bb;
    a.x  = wrow[kb];
    a.y  = wrow[kb + 1];
    bb.x = xb[(size_t)(kb + 0) * 16384];
    bb.y = xb[(size_t)(kb + 1) * 16384];
    acc = WMMA_F32_4(a, bb, acc);
  }

  float* outp = qkv + (size_t)b * 256 * 16384 + hw;
#pragma unroll
  for (int r = 0; r < 8; ++r) {
    const int o = mT * 16 + hl * 8 + r;     // D row for this lane/register
    outp[(size_t)o * 16384] = acc[r];
    float s  = redAdd16(acc[r]);            // sum over the 16 positions
    float q2 = redAdd16(acc[r] * acc[r]);
    if (ln == 0) {                          // lane 0 and lane 16 (diff channels)
      atomicAdd(&ch_sum[o], s);
      atomicAdd(&ch_ss[o], q2);
    }
  }
}

// K2: finalize projection BN (256 channels over N*H*W = 32768 elements).
__global__ void k_fin1(const float* __restrict__ ch_sum, const float* __restrict__ ch_ss,
                       const float* __restrict__ bnq_g, const float* __restrict__ bnq_b,
                       const float* __restrict__ bnk_g, const float* __restrict__ bnk_b,
                       const float* __restrict__ bnv_g, const float* __restrict__ bnv_b,
                       float* __restrict__ sc, float* __restrict__ sh) {
  const int o = threadIdx.x;                 // 256 threads
  const float inv = 1.0f / 32768.0f;
  float m = ch_sum[o] * inv;
  float v = ch_ss[o] * inv - m * m;
  float r = rsqrtf(v + 1e-5f);
  float g, bb;
  if (o < 64)       { g = bnq_g[o];       bb = bnq_b[o]; }
  else if (o < 128) { g = bnk_g[o - 64];  bb = bnk_b[o - 64]; }
  else              { g = bnv_g[o - 128]; bb = bnv_b[o - 128]; }
  const float s = g * r;
  sc[o] = s;
  sh[o] = bb - m * s;
}

// ---------------------------------------------------------------------------
// K3/K5: attention over one (b, g, w) slice. 8 waves; wave `iT` owns i-rows
// [iT*16, iT*16+16). qk via WMMA (K=8), qr/kr via VALU gathers of the rel
// tables (L2-resident). PASS 0 only accumulates group BN stats; PASS 1
// normalizes, softmaxes over j, then sv via WMMA (K=128) and sve via VALU.
// ---------------------------------------------------------------------------
template <int PASS>
__global__ void k_attn(const float* __restrict__ qkv,
                       const float* __restrict__ sc1, const float* __restrict__ sh1,
                       const float* __restrict__ q_rel, const float* __restrict__ k_rel,
                       const float* __restrict__ v_rel,
                       float* __restrict__ gstat,
                       const float* __restrict__ g_sc, const float* __restrict__ g_sh,
                       float* __restrict__ svb, float* __restrict__ sveb,
                       float* __restrict__ sv_sum, float* __restrict__ sv_ss,
                       float* __restrict__ sve_sum, float* __restrict__ sve_ss) {
  extern __shared__ float smem[];
  float* qS   = smem;            // [8][128]   q[c][i]
  float* kS   = qS + 1024;       // [8][128]   k[c][i]
  float* vS   = kS + 1024;       // [128][16]  v transposed: [j][c]  (PASS1)
  float* simS = vS + 2048;       // [128][130] padded rows -> no bank conflicts
  const int SIMP = 130;

  const int b = blockIdx.x >> 10;
  const int g = (blockIdx.x >> 7) & 7;
  const int w = blockIdx.x & 127;

  for (int e = threadIdx.x; e < 1024; e += 256) {
    const int c = e >> 7, i = e & 127;
    const int oq = g * 8 + c;
    const int ok = 64 + g * 8 + c;
    const size_t base = (size_t)b * 256 * 16384 + (size_t)i * 128 + w;
    qS[e] = qkv[base + (size_t)oq * 16384] * sc1[oq] + sh1[oq];
    kS[e] = qkv[base + (size_t)ok * 16384] * sc1[ok] + sh1[ok];
  }
  if constexpr (PASS == 1) {
    for (int e = threadIdx.x; e < 2048; e += 256) {
      const int c = e >> 7, i = e & 127;
      const int ov = 128 + g * 16 + c;
      vS[i * 16 + c] =
          qkv[((size_t)b * 256 + ov) * 16384 + (size_t)i * 128 + w] * sc1[ov] + sh1[ov];
    }
  }
  __syncthreads();

  const int lane = threadIdx.x & 31;
  const int iT   = threadIdx.x >> 5;   // wave id = i-tile
  const int hl   = lane >> 4;
  const int ln   = lane & 15;

  float qksc = 0, qksh = 0, qrsc = 0, qrsh = 0, krsc = 0, krsh = 0;
  if constexpr (PASS == 1) {
    qksc = g_sc[g];      qksh = g_sh[g];
    qrsc = g_sc[8 + g];  qrsh = g_sh[8 + g];
    krsc = g_sc[16 + g]; krsh = g_sh[16 + g];
  }

  float s_qk = 0, ss_qk = 0, s_qr = 0, ss_qr = 0, s_kr = 0, ss_kr = 0;
  v8f L[8];

#pragma unroll
  for (int t = 0; t < 8; ++t) {
    v8f acc = {0.f, 0.f, 0.f, 0.f, 0.f, 0.f, 0.f, 0.f};
#pragma unroll
    for (int kk = 0; kk < 8; kk += 4) {
      const int kb = kk + hl * 2;
      v2f a, bb;
      a.x  = qS[(kb + 0) * 128 + iT * 16 + ln];   // A[m=i, k=c] = q[c][i]
      a.y  = qS[(kb + 1) * 128 + iT * 16 + ln];
      bb.x = kS[(kb + 0) * 128 + t * 16 + ln];    // B[k=c, n=j] = k[c][j]
      bb.y = kS[(kb + 1) * 128 + t * 16 + ln];
      acc = WMMA_F32_4(a, bb, acc);
    }
    const int j = t * 16 + ln;
#pragma unroll
    for (int r = 0; r < 8; ++r) {
      const int i = iT * 16 + hl * 8 + r;
      float qr = 0.f, kr = 0.f;
      const float* qrl = q_rel + (i - j + 127);   // idx in [0,254]
      const float* krl = k_rel + (j - i + 127);
#pragma unroll
      for (int c = 0; c < 8; ++c) {
        qr = fmaf(qS[c * 128 + i], qrl[c * 255], qr);
        kr = fmaf(kS[c * 128 + j], krl[c * 255], kr);   // kr^T term at (i,j)
      }
      if constexpr (PASS == 0) {
        const float qk = acc[r];
        s_qk += qk; ss_qk += qk * qk;
        s_qr += qr; ss_qr += qr * qr;
        s_kr += kr; ss_kr += kr * kr;
      } else {
        L[t][r] = acc[r] * qksc + qksh + qr * qrsc + qrsh + kr * krsc + krsh;
      }
    }
  }

  if constexpr (PASS == 0) {
    s_qk = redAdd32(s_qk);  ss_qk = redAdd32(ss_qk);
    s_qr = redAdd32(s_qr);  ss_qr = redAdd32(ss_qr);
    s_kr = redAdd32(s_kr);  ss_kr = redAdd32(ss_kr);
    if (lane == 0) {
      atomicAdd(&gstat[g * 2 + 0], s_qk);         atomicAdd(&gstat[g * 2 + 1], ss_qk);
      atomicAdd(&gstat[(8 + g) * 2 + 0], s_qr);   atomicAdd(&gstat[(8 + g) * 2 + 1], ss_qr);
      atomicAdd(&gstat[(16 + g) * 2 + 0], s_kr);  atomicAdd(&gstat[(16 + g) * 2 + 1], ss_kr);
    }
    return;
  }

  if constexpr (PASS == 1) {
    // softmax over j (axis spread across 8 j-tiles x 16 lanes)
#pragma unroll
    for (int r = 0; r < 8; ++r) {
      float mx = L[0][r];
#pragma unroll
      for (int t = 1; t < 8; ++t) mx = fmaxf(mx, L[t][r]);
      mx = redMax16(mx);
      float sum = 0.f;
#pragma unroll
      for (int t = 0; t < 8; ++t) {
        const float e = __expf(L[t][r] - mx);
        L[t][r] = e;
        sum += e;
      }
      sum = redAdd16(sum);
      const float inv = 1.0f / sum;
#pragma unroll
      for (int t = 0; t < 8; ++t) L[t][r] *= inv;
    }

    // stage sim rows in LDS (wave-private rows; padded stride)
#pragma unroll
    for (int t = 0; t < 8; ++t)
#pragma unroll
      for (int r = 0; r < 8; ++r)
        simS[(iT * 16 + hl * 8 + r) * SIMP + t * 16 + ln] = L[t][r];
    __syncthreads();

    // sv = sim @ v^T  : M=i (16), N=c (16), K=j (128) on the matrix pipe
    v8f sacc = {0.f, 0.f, 0.f, 0.f, 0.f, 0.f, 0.f, 0.f};
    for (int kk = 0; kk < 128; kk += 4) {
      const int kb = kk + hl * 2;
      v2f a, bb;
      a.x  = simS[(iT * 16 + ln) * SIMP + kb];
      a.y  = simS[(iT * 16 + ln) * SIMP + kb + 1];
      bb.x = vS[(kb + 0) * 16 + ln];             // B[k=j, n=c] = v[c][j]
      bb.y = vS[(kb + 1) * 16 + ln];
      sacc = WMMA_F32_4(a, bb, sacc);
    }

    // sve[c,i] = sum_j sim[i,j] * v_rel[c, j-i+127]  (i-dependent: VALU)
    float eacc[8];
#pragma unroll
    for (int r = 0; r < 8; ++r) {
      const int i = iT * 16 + hl * 8 + r;
      const float* sr = simS + (size_t)i * SIMP;
      const float* vr = v_rel + ln * 255 + (127 - i);
      float av = 0.f;
      for (int j = 0; j < 128; ++j) av = fmaf(sr[j], vr[j], av);
      eacc[r] = av;
    }

    // store sv/sve (matches D lane layout of sacc) + channel stats
    const int Cg = g * 16 + ln;
    float ssum = 0, ssq = 0, esum = 0, esq = 0;
#pragma unroll
    for (int r = 0; r < 8; ++r) {
      const int i = iT * 16 + hl * 8 + r;
      const size_t oidx = (((size_t)b * 128 + Cg) * 128 + i) * 128 + w;
      const float sval = sacc[r];
      const float ev   = eacc[r];
      svb[oidx]  = sval;
      sveb[oidx] = ev;
      ssum += sval; ssq += sval * sval;
      esum += ev;   esq += ev * ev;
    }
    ssum += swzXor<16>(ssum);
    ssq  += swzXor<16>(ssq);
    esum += swzXor<16>(esum);
    esq  += swzXor<16>(esq);
    if (hl == 0) {
      atomicAdd(&sv_sum[Cg], ssum);   atomicAdd(&sv_ss[Cg], ssq);
      atomicAdd(&sve_sum[Cg], esum);  atomicAdd(&sve_ss[Cg], esq);
    }
  }
}

// K4: finalize group BNs for qk/qr/kr (count = N*H*H*W = 4194304 per group).
__global__ void k_fin2(const float* __restrict__ gstat,
                       const float* __restrict__ qk_g, const float* __restrict__ qk_b,
                       const float* __restrict__ qr_g, const float* __restrict__ qr_b,
                       const float* __restrict__ kr_g, const float* __restrict__ kr_b,
                       float* __restrict__ g_sc, float* __restrict__ g_sh) {
  const int t = threadIdx.x;
  if (t >= 24) return;
  const int bn = t >> 3, g = t & 7;
  const float inv = 1.0f / 4194304.0f;
  float m = gstat[t * 2 + 0] * inv;
  float v = gstat[t * 2 + 1] * inv - m * m;
  float r = rsqrtf(v + 1e-5f);
  float gg, bb;
  if (bn == 0)      { gg = qk_g[g]; bb = qk_b[g]; }
  else if (bn == 1) { gg = qr_g[g]; bb = qr_b[g]; }
  else              { gg = kr_g[g]; bb = kr_b[g]; }
  const float s = gg * r;
  g_sc[t] = s;
  g_sh[t] = bb - m * s;
}

// K6: finalize sv/sve BNs (128 channels each, count = 32768).
__global__ void k_fin3(const float* __restrict__ sv_sum, const float* __restrict__ sv_ss,
                       const float* __restrict__ sve_sum, const float* __restrict__ sve_ss,
                       const float* __restrict__ sv_g, const float* __restrict__ sv_b,
                       const float* __restrict__ sve_g, const float* __restrict__ sve_b,
                       float* __restrict__ sv_sc, float* __restrict__ sv_sh,
                       float* __restrict__ sve_sc, float* __restrict__ sve_sh) {
  const int t = threadIdx.x;  // 256
  const float inv = 1.0f / 32768.0f;
  if (t < 128) {
    float m = sv_sum[t] * inv;
    float v = sv_ss[t] * inv - m * m;
    float s = sv_g[t] * rsqrtf(v + 1e-5f);
    sv_sc[t] = s;
    sv_sh[t] = sv_b[t] - m * s;
  } else {
    const int c = t - 128;
    float m = sve_sum[c] * inv;
    float v = sve_ss[c] * inv - m * m;
    float s = sve_g[c] * rsqrtf(v + 1e-5f);
    sve_sc[c] = s;
    sve_sh[c] = sve_b[c] - m * s;
  }
}

// K7: out = bn(sv) + bn(sve), vectorized 4-wide (channel runs are 16384
// elements, so a float4 never straddles a channel boundary).
__global__ void k_final(const float4* __restrict__ svb, const float4* __restrict__ sveb,
                        const float* __restrict__ sv_sc, const float* __restrict__ sv_sh,
                        const float* __restrict__ sve_sc, const float* __restrict__ sve_sh,
                        float4* __restrict__ out) {
  const size_t v = (size_t)blockIdx.x * 256 + threadIdx.x;  // 1048576 float4s
  const int c = (int)((v >> 12) & 127);
  const float4 a = svb[v];
  const float4 e = sveb[v];
  const float s1 = sv_sc[c], h1 = sv_sh[c];
  const float s2 = sve_sc[c], h2 = sve_sh[c];
  float4 o;
  o.x = a.x * s1 + h1 + e.x * s2 + h2;
  o.y = a.y * s1 + h1 + e.y * s2 + h2;
  o.z = a.z * s1 + h1 + e.z * s2 + h2;
  o.w = a.w * s1 + h1 + e.w * s2 + h2;
  out[v] = o;
}

// ---------------------------------------------------------------------------
extern "C" void kernel_launch(void* const* d_in, const int* in_sizes, int n_in,
                              void* d_out, int out_size, void* d_ws, size_t ws_size,
                              hipStream_t stream) {
  (void)in_sizes; (void)n_in; (void)out_size; (void)ws_size;
  const float* x     = (const float*)d_in[0];
  const float* wq    = (const float*)d_in[1];
  const float* wk    = (const float*)d_in[2];
  const float* wv    = (const float*)d_in[3];
  const float* q_rel = (const float*)d_in[4];
  const float* k_rel = (const float*)d_in[5];
  const float* v_rel = (const float*)d_in[6];
  const float* bnq_g  = (const float*)d_in[7],  * bnq_b  = (const float*)d_in[8];
  const float* bnk_g  = (const float*)d_in[9],  * bnk_b  = (const float*)d_in[10];
  const float* bnv_g  = (const float*)d_in[11], * bnv_b  = (const float*)d_in[12];
  const float* bnqk_g = (const float*)d_in[13], * bnqk_b = (const float*)d_in[14];
  const float* bnqr_g = (const float*)d_in[15], * bnqr_b = (const float*)d_in[16];
  const float* bnkr_g = (const float*)d_in[17], * bnkr_b = (const float*)d_in[18];
  const float* bnsv_g = (const float*)d_in[19], * bnsv_b = (const float*)d_in[20];
  const float* bnsve_g = (const float*)d_in[21], * bnsve_b = (const float*)d_in[22];
  float* out = (float*)d_out;

  float* ws   = (float*)d_ws;
  float* qkv  = ws;                   // 8388608 floats
  float* svb  = ws + 8388608;         // 4194304
  float* sveb = ws + 12582912;        // 4194304
  float* st   = ws + 16777216;        // stats block (2144 floats)
  float* ch_sum = st;        float* ch_ss = st + 256;
  float* ch_sc  = st + 512;  float* ch_sh = st + 768;
  float* gstat  = st + 1024;                         // [3][8][2]
  float* g_sc   = st + 1072; float* g_sh  = st + 1096;
  float* sv_sum = st + 1120; float* sv_ss = st + 1248;
  float* sve_sum = st + 1376; float* sve_ss = st + 1504;
  float* sv_sc  = st + 1632; float* sv_sh = st + 1760;
  float* sve_sc = st + 1888; float* sve_sh = st + 2016;

  hipMemsetAsync(st, 0, 2144 * sizeof(float), stream);

  // PASS1 needs 82944B of LDS (fits WGP's 320KB, above the 64KB default cap)
  (void)hipFuncSetAttribute(reinterpret_cast<const void*>(k_attn<1>),
                            hipFuncAttributeMaxDynamicSharedMemorySize, 82944);

  k_gemm_qkv<<<4096, 256, 0, stream>>>(x, wq, wk, wv, qkv, ch_sum, ch_ss);
  k_fin1<<<1, 256, 0, stream>>>(ch_sum, ch_ss, bnq_g, bnq_b, bnk_g, bnk_b,
                                bnv_g, bnv_b, ch_sc, ch_sh);
  k_attn<0><<<2048, 256, 8192, stream>>>(qkv, ch_sc, ch_sh, q_rel, k_rel, v_rel,
                                         gstat, nullptr, nullptr,
                                         nullptr, nullptr, nullptr, nullptr,
                                         nullptr, nullptr);
  k_fin2<<<1, 32, 0, stream>>>(gstat, bnqk_g, bnqk_b, bnqr_g, bnqr_b,
                               bnkr_g, bnkr_b, g_sc, g_sh);
  k_attn<1><<<2048, 256, 82944, stream>>>(qkv, ch_sc, ch_sh, q_rel, k_rel, v_rel,
                                          nullptr, g_sc, g_sh,
                                          svb, sveb, sv_sum, sv_ss,
                                          sve_sum, sve_ss);
  k_fin3<<<1, 256, 0, stream>>>(sv_sum, sv_ss, sve_sum, sve_ss,
                                bnsv_g, bnsv_b, bnsve_g, bnsve_b,
                                sv_sc, sv_sh, sve_sc, sve_sh);
  k_final<<<4096, 256, 0, stream>>>((const float4*)svb, (const float4*)sveb,
                                    sv_sc, sv_sh, sve_sc, sve_sh, (float4*)out);
}